// ResNet_Grad_83485574299808
// MI455X (gfx1250) — compile-verified
//
#include <hip/hip_runtime.h>
#include <hip/hip_bf16.h>

#define NPIX 3136       // 56*56 = N = M
#define HW   56
#define MSPLIT 8
#define MCHUNK (NPIX / MSPLIT)   // 392 rows per chunk, 98 wmma steps of K=4

typedef float v2f __attribute__((ext_vector_type(2)));
typedef float v8f __attribute__((ext_vector_type(8)));

// ---------------------------------------------------------------------------
// Kernel 1: r[m] = sum_n A[m,n]   (row sums; coalesced streaming read of A)
// ---------------------------------------------------------------------------
__global__ __launch_bounds__(128) void rowsum_kernel(const float* __restrict__ A,
                                                     float* __restrict__ r) {
    __shared__ float red[128];
    const int m = blockIdx.x;
    const float* row = A + (size_t)m * NPIX;
    float s = 0.0f;
    for (int j = threadIdx.x; j < NPIX; j += 128) s += row[j];
    red[threadIdx.x] = s;
    __syncthreads();
    for (int st = 64; st > 0; st >>= 1) {
        if (threadIdx.x < st) red[threadIdx.x] += red[threadIdx.x + st];
        __syncthreads();
    }
    if (threadIdx.x == 0) r[m] = red[0];
}

// ---------------------------------------------------------------------------
// Kernel 2: partial[chunk][n] = sum_{m in chunk} r[m] * A[m,n]
// via V_WMMA_F32_16X16X4_F32:  D(16x16) = Aop(16x4) * Bop(4x16) + C
//   Bop[k][j] = A[m+k][n0+j]   (4 rows of A, 16 output columns)
//   Aop[i][k] = r[m+k]         (replicated across all 16 rows)
// ISA 32-bit layouts (wave32): A 16x4: lanes0-15 {K0,K1} in v0,v1; lanes16-31
// {K2,K3}.  B 4x16 mirrors with N across lanes.  Same slot->K indexing is
// used for both operands, so the pairing r[m+k]*A[m+k][col] is exact fp32.
// ---------------------------------------------------------------------------
__global__ __launch_bounds__(32) void atr_wmma_kernel(const float* __restrict__ A,
                                                      const float* __restrict__ r,
                                                      float* __restrict__ partial) {
    const int lane = threadIdx.x;            // full wave, EXEC all ones for WMMA
    const int col  = blockIdx.x * 16 + (lane & 15);
    const int half = lane >> 4;              // selects K pair {0,1} vs {2,3}
    const int mbase = blockIdx.y * MCHUNK;

    v8f c = {0.f, 0.f, 0.f, 0.f, 0.f, 0.f, 0.f, 0.f};
    for (int i = 0; i < MCHUNK / 4; ++i) {
        const int m = mbase + i * 4 + 2 * half;
        v2f a, b;
        a.x = r[m + 0];
        a.y = r[m + 1];
        const float* ap = A + (size_t)m * NPIX + col;
        b.x = ap[0];
        b.y = ap[NPIX];
        c = __builtin_amdgcn_wmma_f32_16x16x4_f32(
                /*neg_a=*/false, a, /*neg_b=*/false, b,
                /*c_mod=*/(short)0, c, /*reuse_a=*/false, /*reuse_b=*/false);
    }
    // D row M=0 lives in VGPR0 on lanes 0..15 (all rows identical by design)
    if (lane < 16) partial[(size_t)blockIdx.y * NPIX + col] = c[0];
}

// ---------------------------------------------------------------------------
// Kernel 3: gr[n] = (1/N) * sum_chunks partial[chunk][n]   (deterministic)
// ---------------------------------------------------------------------------
__global__ __launch_bounds__(256) void combine_kernel(const float* __restrict__ partial,
                                                      float* __restrict__ gr) {
    const int n = blockIdx.x * 256 + threadIdx.x;
    if (n < NPIX) {
        float s = 0.0f;
        for (int k = 0; k < MSPLIT; ++k) s += partial[(size_t)k * NPIX + n];
        gr[n] = s * (1.0f / (float)NPIX);
    }
}

// ---------------------------------------------------------------------------
// Kernel 4: the 4 unrolled ResBlocks. gr is constant across layers. One
// workgroup, all tensors LDS-resident, block barriers between conv/BN stages.
// ---------------------------------------------------------------------------
__global__ __launch_bounds__(1024) void resnet_kernel(
        const float* __restrict__ f_in, const float* __restrict__ gr,
        const float* __restrict__ c1w, const float* __restrict__ c1b,
        const float* __restrict__ b1g, const float* __restrict__ b1b,
        const float* __restrict__ c2w, const float* __restrict__ c2b,
        const float* __restrict__ b2g, const float* __restrict__ b2b,
        const float* __restrict__ pw,  const float* __restrict__ pb,
        float* __restrict__ out) {
    __shared__ float fb[NPIX], gb[NPIX], t1[NPIX], t2[NPIX];
    __shared__ float red1[1024], red2[1024];
    const int tid = threadIdx.x;

    for (int p = tid; p < NPIX; p += 1024) { fb[p] = f_in[p]; gb[p] = gr[p]; }
    __syncthreads();

    for (int L = 0; L < 4; ++L) {
        // ---- conv1: 2->1 ch, 3x3, pad 1 (cross-correlation, OIHW) ----
        const float* w1 = c1w + L * 18;
        for (int p = tid; p < NPIX; p += 1024) {
            const int y = p / HW, x = p % HW;
            float acc = c1b[L];
            for (int ky = 0; ky < 3; ++ky) {
                const int yy = y + ky - 1;
                if (yy < 0 || yy >= HW) continue;
                for (int kx = 0; kx < 3; ++kx) {
                    const int xx = x + kx - 1;
                    if (xx < 0 || xx >= HW) continue;
                    const int q = yy * HW + xx;
                    acc += w1[ky * 3 + kx] * fb[q] + w1[9 + ky * 3 + kx] * gb[q];
                }
            }
            t1[p] = acc;
        }
        __syncthreads();
        // ---- BN1 (training mode, biased var) + ReLU ----
        float s = 0.0f, ss = 0.0f;
        for (int p = tid; p < NPIX; p += 1024) { const float v = t1[p]; s += v; ss += v * v; }
        red1[tid] = s; red2[tid] = ss; __syncthreads();
        for (int st = 512; st > 0; st >>= 1) {
            if (tid < st) { red1[tid] += red1[tid + st]; red2[tid] += red2[tid + st]; }
            __syncthreads();
        }
        {
            const float mu  = red1[0] * (1.0f / NPIX);
            const float var = red2[0] * (1.0f / NPIX) - mu * mu;
            const float rs  = rsqrtf(var + 1e-5f);
            const float ga = b1g[L], be = b1b[L];
            for (int p = tid; p < NPIX; p += 1024)
                t1[p] = fmaxf(ga * ((t1[p] - mu) * rs) + be, 0.0f);
        }
        __syncthreads();
        // ---- conv2: 1->1 ch, 3x3, pad 1 ----
        const float* w2 = c2w + L * 9;
        for (int p = tid; p < NPIX; p += 1024) {
            const int y = p / HW, x = p % HW;
            float acc = c2b[L];
            for (int ky = 0; ky < 3; ++ky) {
                const int yy = y + ky - 1;
                if (yy < 0 || yy >= HW) continue;
                for (int kx = 0; kx < 3; ++kx) {
                    const int xx = x + kx - 1;
                    if (xx < 0 || xx >= HW) continue;
                    acc += w2[ky * 3 + kx] * t1[yy * HW + xx];
                }
            }
            t2[p] = acc;
        }
        __syncthreads();
        // ---- BN2 + 1x1 projection of [f, gr] + residual ReLU ----
        s = 0.0f; ss = 0.0f;
        for (int p = tid; p < NPIX; p += 1024) { const float v = t2[p]; s += v; ss += v * v; }
        red1[tid] = s; red2[tid] = ss; __syncthreads();
        for (int st = 512; st > 0; st >>= 1) {
            if (tid < st) { red1[tid] += red1[tid + st]; red2[tid] += red2[tid + st]; }
            __syncthreads();
        }
        {
            const float mu  = red1[0] * (1.0f / NPIX);
            const float var = red2[0] * (1.0f / NPIX) - mu * mu;
            const float rs  = rsqrtf(var + 1e-5f);
            const float ga = b2g[L], be = b2b[L];
            const float p0 = pw[L * 2 + 0], p1 = pw[L * 2 + 1], pbb = pb[L];
            for (int p = tid; p < NPIX; p += 1024) {
                const float idn = p0 * fb[p] + p1 * gb[p] + pbb;   // own-pixel read/write: safe
                const float v   = ga * ((t2[p] - mu) * rs) + be + idn;
                fb[p] = fmaxf(v, 0.0f);
            }
        }
        __syncthreads();   // fb stable before next layer's 3x3 neighborhood reads
    }
    for (int p = tid; p < NPIX; p += 1024) out[p] = fb[p];
}

// ---------------------------------------------------------------------------
extern "C" void kernel_launch(void* const* d_in, const int* in_sizes, int n_in,
                              void* d_out, int out_size, void* d_ws, size_t ws_size,
                              hipStream_t stream) {
    const float* f   = (const float*)d_in[0];
    // d_in[1] (g) is mathematically irrelevant: the grad-block loss is linear
    // in f, so its gradient is (1/N)*A^T(A*1), independent of f and g.
    const float* A   = (const float*)d_in[2];
    const float* c1w = (const float*)d_in[3];
    const float* c1b = (const float*)d_in[4];
    const float* b1g = (const float*)d_in[5];
    const float* b1b = (const float*)d_in[6];
    const float* c2w = (const float*)d_in[7];
    const float* c2b = (const float*)d_in[8];
    const float* b2g = (const float*)d_in[9];
    const float* b2b = (const float*)d_in[10];
    const float* pw  = (const float*)d_in[11];
    const float* pb  = (const float*)d_in[12];

    float* ws      = (float*)d_ws;
    float* r       = ws;                       // NPIX
    float* partial = ws + NPIX;                // MSPLIT * NPIX
    float* gr      = ws + NPIX + MSPLIT * NPIX;// NPIX

    rowsum_kernel<<<NPIX, 128, 0, stream>>>(A, r);
    atr_wmma_kernel<<<dim3(NPIX / 16, MSPLIT), 32, 0, stream>>>(A, r, partial);
    combine_kernel<<<(NPIX + 255) / 256, 256, 0, stream>>>(partial, gr);
    resnet_kernel<<<1, 1024, 0, stream>>>(f, gr, c1w, c1b, b1g, b1b,
                                          c2w, c2b, b2g, b2b, pw, pb,
                                          (float*)d_out);
}